// GNN_3813930959125
// MI455X (gfx1250) — compile-verified
//
#include <hip/hip_runtime.h>

// ---------------------------------------------------------------------------
// 2-layer GCN for MI455X (gfx1250, wave32).
// All dense 16-wide GEMMs use V_WMMA_F32_16X16X4_F32 chained over K.
// ew = edge_attr@We is recomputed in every edge pass (205MB read beats
// 205MB write + 2x205MB read; node arrays are L2-resident at 6.4MB each).
// Edge waves own 4 consecutive 16-edge tiles (4KB of edge_attr) with the
// We fragments hoisted, plus a global_prefetch of the wave's chunk.
// ---------------------------------------------------------------------------

typedef __attribute__((ext_vector_type(2))) float v2f;
typedef __attribute__((ext_vector_type(8))) float v8f;

#define N_NODES 100000
#define E_EDGES 3200000
#define IN_CH   128
#define C16     16
#define TPW     4          // tiles per wave in edge kernels

// ---- WMMA f32 16x16x4 helpers ---------------------------------------------
// A (16x4, f32): lane L<16 holds A[L][k0],A[L][k1]; lane L>=16 holds A[L-16][k2],A[L-16][k3]
// B (4x16, f32): lane L<16 holds B[k0][L],B[k1][L]; lane L>=16 holds B[k2][L-16],B[k3][L-16]
// C/D (16x16):   vgpr r = row (r + 8*(lane>=16)), col = lane&15

__device__ __forceinline__ v8f wmma_f32_16x16x4(v2f a, v2f b, v8f c) {
  return __builtin_amdgcn_wmma_f32_16x16x4_f32(
      /*neg_a=*/false, a, /*neg_b=*/false, b,
      /*c_mod=*/(short)0, c, /*reuse_a=*/false, /*reuse_b=*/false);
}

__device__ __forceinline__ v2f load_a_frag(const float* __restrict__ A,
                                           int tileRow, int lda, int kstep, int lane) {
  int row = tileRow + (lane & 15);
  int kb  = (kstep << 2) + ((lane >> 4) << 1);   // even -> 8B aligned
  return *(const v2f*)(A + (size_t)row * lda + kb);
}

__device__ __forceinline__ v2f load_b_frag(const float* __restrict__ B,
                                           int kstep, int lane) {
  int col = lane & 15;
  int kb  = (kstep << 2) + ((lane >> 4) << 1);
  v2f b;
  b.x = B[kb * C16 + col];
  b.y = B[(kb + 1) * C16 + col];
  return b;
}

// ---- Elementwise kernels ---------------------------------------------------

__global__ void __launch_bounds__(256) fill_one_kernel(float* __restrict__ p) {
  p[blockIdx.x * 256 + threadIdx.x] = 1.0f;
}

__global__ void __launch_bounds__(256) dinv_kernel(float* __restrict__ deg) {
  int i = blockIdx.x * 256 + threadIdx.x;
  float d = deg[i];
  deg[i] = (d > 0.0f) ? rsqrtf(fabsf(d) + 1e-12f) : 0.0f;
}

// u = dinv * t ; agg = dinv^2 * t  (self-loop contribution pre-seeded)
__global__ void __launch_bounds__(256) prep_kernel(const float* __restrict__ dinv,
                                                   const float* __restrict__ t,
                                                   float* __restrict__ u,
                                                   float* __restrict__ agg) {
  int i = blockIdx.x * 256 + threadIdx.x;
  float dv = dinv[i], tv = t[i];
  float uv = dv * tv;
  u[i]   = uv;
  agg[i] = dv * uv;
}

__global__ void __launch_bounds__(256) finalize_kernel(const float* __restrict__ agg,
                                                       const float* __restrict__ bias,
                                                       float* __restrict__ outp,
                                                       int do_relu) {
  int i = blockIdx.x * 256 + threadIdx.x;
  float v = agg[i] + bias[i & 15];
  if (do_relu) v = fmaxf(v, 0.0f);
  outp[i] = v;
}

// ---- WMMA edge kernels -----------------------------------------------------

// deg[dst[e], c] += (edge_attr @ We + be)[e, c]
// One wave per TPW consecutive 16-edge tiles; We fragments hoisted.
__global__ void __launch_bounds__(256) edge_deg_kernel(
    const float* __restrict__ edge_attr, const int* __restrict__ edge_index,
    const float* __restrict__ We, const float* __restrict__ be,
    float* __restrict__ deg) {
  int lane  = threadIdx.x & 31;
  int wave  = blockIdx.x * 8 + (threadIdx.x >> 5);
  int tile0 = wave * TPW;
  const int* __restrict__ dstI = edge_index + E_EDGES;

  int col  = lane & 15;
  int half = lane >> 4;
  float bev = be[col];
  v2f b0 = load_b_frag(We, 0, lane);
  v2f b1 = load_b_frag(We, 1, lane);
  v2f b2 = load_b_frag(We, 2, lane);
  v2f b3 = load_b_frag(We, 3, lane);

  // Pull this wave's 4KB edge_attr chunk toward the caches (128B per lane).
  __builtin_prefetch(edge_attr + ((size_t)tile0 << 4) * C16 + lane * 32, 0, 3);

  for (int t = 0; t < TPW; ++t) {              // wave-uniform loop
    int tileRow = (tile0 + t) << 4;
    v8f c = {};
    c = wmma_f32_16x16x4(load_a_frag(edge_attr, tileRow, C16, 0, lane), b0, c);
    c = wmma_f32_16x16x4(load_a_frag(edge_attr, tileRow, C16, 1, lane), b1, c);
    c = wmma_f32_16x16x4(load_a_frag(edge_attr, tileRow, C16, 2, lane), b2, c);
    c = wmma_f32_16x16x4(load_a_frag(edge_attr, tileRow, C16, 3, lane), b3, c);
#pragma unroll
    for (int r = 0; r < 8; ++r) {
      int erow = tileRow + r + (half << 3);
      int d    = dstI[erow];
      atomicAdd(&deg[d * C16 + col], c[r] + bev);
    }
  }
}

// agg[dst,c] += ew[e,c] * u[src,c] * dinv[dst,c]   with ew recomputed via WMMA
__global__ void __launch_bounds__(256) edge_agg_kernel(
    const float* __restrict__ edge_attr, const int* __restrict__ edge_index,
    const float* __restrict__ We, const float* __restrict__ be,
    const float* __restrict__ u, const float* __restrict__ dinv,
    float* __restrict__ agg) {
  int lane  = threadIdx.x & 31;
  int wave  = blockIdx.x * 8 + (threadIdx.x >> 5);
  int tile0 = wave * TPW;
  const int* __restrict__ srcI = edge_index;
  const int* __restrict__ dstI = edge_index + E_EDGES;

  int col  = lane & 15;
  int half = lane >> 4;
  float bev = be[col];
  v2f b0 = load_b_frag(We, 0, lane);
  v2f b1 = load_b_frag(We, 1, lane);
  v2f b2 = load_b_frag(We, 2, lane);
  v2f b3 = load_b_frag(We, 3, lane);

  __builtin_prefetch(edge_attr + ((size_t)tile0 << 4) * C16 + lane * 32, 0, 3);

  for (int t = 0; t < TPW; ++t) {              // wave-uniform loop
    int tileRow = (tile0 + t) << 4;
    v8f c = {};
    c = wmma_f32_16x16x4(load_a_frag(edge_attr, tileRow, C16, 0, lane), b0, c);
    c = wmma_f32_16x16x4(load_a_frag(edge_attr, tileRow, C16, 1, lane), b1, c);
    c = wmma_f32_16x16x4(load_a_frag(edge_attr, tileRow, C16, 2, lane), b2, c);
    c = wmma_f32_16x16x4(load_a_frag(edge_attr, tileRow, C16, 3, lane), b3, c);
#pragma unroll
    for (int r = 0; r < 8; ++r) {
      int erow = tileRow + r + (half << 3);
      int sN   = srcI[erow];
      int dN   = dstI[erow];
      float ew = c[r] + bev;
      float m  = ew * u[sN * C16 + col] * dinv[dN * C16 + col];
      atomicAdd(&agg[dN * C16 + col], m);
    }
  }
}

// T = X @ W   (X: [N, K] row-major, W: [K, 16], T: [N, 16]).
// Wave-stride tile loop with all W fragments hoisted into registers.
template <int K>
__global__ void __launch_bounds__(256) xform_kernel(const float* __restrict__ X,
                                                    const float* __restrict__ W,
                                                    float* __restrict__ T,
                                                    int ntiles) {
  int lane   = threadIdx.x & 31;
  int wave   = blockIdx.x * 8 + (threadIdx.x >> 5);
  int nwaves = gridDim.x * 8;
  int col    = lane & 15;
  int half   = lane >> 4;

  v2f bf[K / 4];
#pragma unroll
  for (int s = 0; s < K / 4; ++s) bf[s] = load_b_frag(W, s, lane);

  for (int tile = wave; tile < ntiles; tile += nwaves) {   // wave-uniform
    int tileRow = tile << 4;
    v8f c = {};
#pragma unroll
    for (int s = 0; s < K / 4; ++s)
      c = wmma_f32_16x16x4(load_a_frag(X, tileRow, K, s, lane), bf[s], c);
#pragma unroll
    for (int r = 0; r < 8; ++r) {
      int row = tileRow + r + (half << 3);
      T[row * C16 + col] = c[r];
    }
  }
}

// ---------------------------------------------------------------------------

extern "C" void kernel_launch(void* const* d_in, const int* in_sizes, int n_in,
                              void* d_out, int out_size, void* d_ws, size_t ws_size,
                              hipStream_t stream) {
  const float* x          = (const float*)d_in[0];
  const int*   edge_index = (const int*)  d_in[1];
  const float* edge_attr  = (const float*)d_in[2];
  const float* We         = (const float*)d_in[3];
  const float* be         = (const float*)d_in[4];
  const float* W1         = (const float*)d_in[5];
  const float* b1         = (const float*)d_in[6];
  const float* W2         = (const float*)d_in[7];
  const float* b2         = (const float*)d_in[8];
  float* out = (float*)d_out;

  const size_t F = (size_t)N_NODES * C16;      // 1.6M floats
  float* dinv = (float*)d_ws;                  // deg -> dinv in place
  float* t    = dinv + F;                      // transformed features
  float* u    = t + F;                         // dinv * t
  float* agg  = u + F;                         // aggregation buffer
  float* h    = agg + F;                       // layer-1 activations

  const int ELEM_BLOCKS = (int)(F / 256);              // 6250 (exact)
  const int EDGE_BLOCKS = E_EDGES / C16 / 8 / TPW;     // 6250 (exact)
  const int NTILES      = N_NODES / C16;               // 6250 (exact)
  const int XT_BLOCKS   = 196;                         // ~4 tiles per wave

  // deg = segment_sum(ew, dst) + 1 ; dinv = rsqrt
  fill_one_kernel<<<ELEM_BLOCKS, 256, 0, stream>>>(dinv);
  edge_deg_kernel<<<EDGE_BLOCKS, 256, 0, stream>>>(edge_attr, edge_index, We, be, dinv);
  dinv_kernel<<<ELEM_BLOCKS, 256, 0, stream>>>(dinv);

  // ---- layer 1: h = relu(gcn_conv(x, W1) + b1)
  xform_kernel<IN_CH><<<XT_BLOCKS, 256, 0, stream>>>(x, W1, t, NTILES);
  prep_kernel<<<ELEM_BLOCKS, 256, 0, stream>>>(dinv, t, u, agg);
  edge_agg_kernel<<<EDGE_BLOCKS, 256, 0, stream>>>(edge_attr, edge_index, We, be,
                                                   u, dinv, agg);
  finalize_kernel<<<ELEM_BLOCKS, 256, 0, stream>>>(agg, b1, h, 1);

  // ---- layer 2: out = gcn_conv(h, W2) + b2
  xform_kernel<C16><<<XT_BLOCKS, 256, 0, stream>>>(h, W2, t, NTILES);
  prep_kernel<<<ELEM_BLOCKS, 256, 0, stream>>>(dinv, t, u, agg);
  edge_agg_kernel<<<EDGE_BLOCKS, 256, 0, stream>>>(edge_attr, edge_index, We, be,
                                                   u, dinv, agg);
  finalize_kernel<<<ELEM_BLOCKS, 256, 0, stream>>>(agg, b2, out, 0);
}